// dec_contrast_78580721648167
// MI455X (gfx1250) — compile-verified
//
#include <hip/hip_runtime.h>
#include <hip/hip_bf16.h>
#include <math.h>

// ---------------- problem constants ----------------
#define BS   8
#define CH   256
#define HW   16384          // 128*128
#define NCLS 19
#define NQ   2975
#define INV_T 5.0f          // 1/0.2

typedef __attribute__((ext_vector_type(2))) float v2f;
typedef __attribute__((ext_vector_type(8))) float v8f;

// ---------------- ws layout (bytes) ----------------
// pred      int [131072]      @ 0         (512 KB)
// sums32    f32 [32*256]      @ 524288    (32 KB)   -- classes padded 19->32
// counts    int [32]          @ 557056
// per_class f32 [32]          @ 557184
// keys      f32 [19*256]      @ 557312
// qsum      f32 [256*2975]    @ 576768    (~3 MB)

__global__ void zero_kernel(float* __restrict__ p, int n) {
  int i = blockIdx.x * blockDim.x + threadIdx.x;
  if (i < n) p[i] = 0.0f;
}

// pred = argmax_k res[b,k,hw]; counts[k] via LDS histogram
__global__ void __launch_bounds__(256)
argmax_kernel(const float* __restrict__ res, int* __restrict__ pred,
              int* __restrict__ counts) {
  __shared__ int hist[NCLS];
  int tid = threadIdx.x;
  if (tid < NCLS) hist[tid] = 0;
  __syncthreads();
  int pix = blockIdx.x * 256 + tid;              // [0, 131072)
  int b   = pix >> 14;
  int hw  = pix & (HW - 1);
  const float* r = res + (size_t)b * NCLS * HW + hw;
  float best = r[0];
  int bk = 0;
#pragma unroll
  for (int k = 1; k < NCLS; ++k) {
    float v = r[(size_t)k * HW];
    if (v > best) { best = v; bk = k; }          // first-max wins (strict >)
  }
  pred[pix] = bk;
  atomicAdd(&hist[bk], 1);
  __syncthreads();
  if (tid < NCLS) atomicAdd(&counts[tid], hist[tid]);
}

// ----------------------------------------------------------------------
// sums[k,c] = sum_p onehot(pred[p]==k) * fea[c,p]  via V_WMMA_F32_16X16X4_F32.
// M = class (2 tiles of 16), N = channel, K = pixels (4 per WMMA).
// fea tiles are staged into double-buffered LDS with GLOBAL_LOAD_ASYNC_TO_LDS
// (ASYNCcnt) so the next tile's DMA overlaps WMMA compute on the current one.
// ----------------------------------------------------------------------
#define TP    16              // pixels per LDS tile
#define TILES 32              // tiles/block -> 512 pixels per block
#define CHUNK (TP * TILES)    // 512

// one tile = 256ch x 16px f32 = 16 KB = 4 async b128 transfers per thread
__device__ __forceinline__ void issue_tile_async(const float* feab, int p0,
                                                 unsigned lds_base, int tid) {
#pragma unroll
  for (int i = 0; i < 4; ++i) {
    int j  = i * 256 + tid;                      // [0, 1024)
    int c  = j >> 2;
    int ps = (j & 3) << 2;
    unsigned lds_addr = lds_base + (unsigned)((c * TP + ps) * 4);
    unsigned goff     = (unsigned)((c * HW + p0 + ps) * 4);
    asm volatile("global_load_async_to_lds_b128 %0, %1, %2 offset:0"
                 :: "v"(lds_addr), "v"(goff), "s"(feab)
                 : "memory");
  }
}

__device__ __forceinline__ void wait_async0() {
  asm volatile("s_wait_asynccnt 0x0" ::: "memory");
}

__global__ void __launch_bounds__(256)
pool_wmma_kernel(const float* __restrict__ fea, const int* __restrict__ pred,
                 float* __restrict__ sums32) {
  __shared__ __align__(16) float sfea[2][CH * TP];  // 2 x 16 KB, [c][p]
  __shared__ __align__(16) int   spred[2][TP];

  int tid  = threadIdx.x;
  int wave = tid >> 5;
  int lane = tid & 31;
  int hi   = lane >> 4;      // lane group: selects K 0/1 vs 2/3 per ISA layout
  int lm   = lane & 15;

  int b      = blockIdx.x >> 5;                  // 32 chunks per batch
  int p0base = (blockIdx.x & 31) * CHUNK;
  const float* feab  = fea  + (size_t)b * CH * HW;
  const int*   predb = pred + (size_t)b * HW;

  // low 32 bits of a flat shared pointer == LDS byte address (ISA aperture map)
  unsigned lds_buf[2];
  lds_buf[0] = (unsigned)(size_t)(&sfea[0][0]);
  lds_buf[1] = (unsigned)(size_t)(&sfea[1][0]);

  v8f acc[2][2];                                 // [m-tile][n-tile]
#pragma unroll
  for (int mt = 0; mt < 2; ++mt)
#pragma unroll
    for (int nt = 0; nt < 2; ++nt)
#pragma unroll
      for (int r = 0; r < 8; ++r) acc[mt][nt][r] = 0.0f;

  int cb = wave * 32;                            // 32 channels per wave
  bool havepred = (tid < TP);

  // prologue: tile 0 into buffer 0
  issue_tile_async(feab, p0base, lds_buf[0], tid);
  if (havepred) spred[0][tid] = predb[p0base + tid];
  wait_async0();
  __syncthreads();

  for (int t = 0; t < TILES; ++t) {
    int buf = t & 1;
    int preg = 0;
    if (t + 1 < TILES) {                         // overlap next tile's DMA
      issue_tile_async(feab, p0base + (t + 1) * TP, lds_buf[(t + 1) & 1], tid);
      if (havepred) preg = predb[p0base + (t + 1) * TP + tid];
    }

#pragma unroll
    for (int kk = 0; kk < TP; kk += 4) {
      // A fragment (one-hot, exact): lane holds K = 2*hi, 2*hi+1
      int2 pp = *(const int2*)(&spred[buf][kk + 2 * hi]);
      v2f a0, a1;
      a0.x = (pp.x == lm)        ? 1.0f : 0.0f;
      a0.y = (pp.y == lm)        ? 1.0f : 0.0f;
      a1.x = (pp.x == (lm + 16)) ? 1.0f : 0.0f;
      a1.y = (pp.y == (lm + 16)) ? 1.0f : 0.0f;
#pragma unroll
      for (int nt = 0; nt < 2; ++nt) {
        int c = cb + nt * 16 + lm;
        float2 bv = *(const float2*)(&sfea[buf][c * TP + kk + 2 * hi]);
        v2f bf; bf.x = bv.x; bf.y = bv.y;
        acc[0][nt] = __builtin_amdgcn_wmma_f32_16x16x4_f32(
            false, a0, false, bf, (short)0, acc[0][nt], false, false);
        acc[1][nt] = __builtin_amdgcn_wmma_f32_16x16x4_f32(
            false, a1, false, bf, (short)0, acc[1][nt], false, false);
      }
    }

    wait_async0();                               // next tile landed in LDS
    if (t + 1 < TILES && havepred) spred[(t + 1) & 1][tid] = preg;
    __syncthreads();                             // all waves done: swap buffers
  }

  // D layout: VGPR r, lanes<16 -> M=r, lanes>=16 -> M=r+8; N = lane%16
#pragma unroll
  for (int mt = 0; mt < 2; ++mt)
#pragma unroll
    for (int nt = 0; nt < 2; ++nt)
#pragma unroll
      for (int r = 0; r < 8; ++r) {
        int m = mt * 16 + hi * 8 + r;
        int c = cb + nt * 16 + lm;
        atomicAdd(&sums32[m * CH + c], acc[mt][nt][r]);
      }
}

// keys = normalize(sums / max(counts,1)) along C
__global__ void __launch_bounds__(256)
keys_kernel(const float* __restrict__ sums32, const int* __restrict__ counts,
            float* __restrict__ keys) {
  __shared__ float red[256];
  int tid = threadIdx.x;
  for (int k = 0; k < NCLS; ++k) {
    float cnt = fmaxf((float)counts[k], 1.0f);
    float v = sums32[k * CH + tid] / cnt;
    red[tid] = v * v;
    __syncthreads();
    for (int s = 128; s > 0; s >>= 1) {
      if (tid < s) red[tid] += red[tid + s];
      __syncthreads();
    }
    float norm = fmaxf(sqrtf(red[0]), 1e-12f);
    __syncthreads();                 // red[0] consumed before next overwrite
    keys[k * CH + tid] = v / norm;
  }
}

// qsum[c*NQ+q] = sum_k queues[k][c*NQ+q]
__global__ void qsum_kernel(const float* __restrict__ queues,
                            float* __restrict__ qsum) {
  int idx = blockIdx.x * blockDim.x + threadIdx.x;
  if (idx >= CH * NQ) return;
  float s = 0.0f;
#pragma unroll
  for (int k = 0; k < NCLS; ++k) s += queues[(size_t)k * CH * NQ + idx];
  qsum[idx] = s;
}

// one wave per (k,c) row: two-pass logsumexp over 2*NQ logits
__global__ void __launch_bounds__(256)
lse_kernel(const float* __restrict__ queues, const float* __restrict__ qsum,
           const float* __restrict__ keys, float* __restrict__ per_class) {
  int tid  = threadIdx.x;
  int wave = tid >> 5, lane = tid & 31;
  int row = blockIdx.x * 8 + wave;               // [0, 19*256)
  if (row >= NCLS * CH) return;
  int k = row / CH, c = row % CH;
  float key = keys[row];
  const float* qrow = queues + (size_t)row * NQ;
  const float* srow = qsum   + (size_t)c   * NQ;

  float m = -INFINITY;
  for (int q = lane; q < NQ; q += 32) {
    float qv = qrow[q], sv = srow[q];
    float lp = key * qv * INV_T;
    float ln = key * (sv - qv) * INV_T;
    m = fmaxf(m, fmaxf(lp, ln));
  }
#pragma unroll
  for (int off = 16; off > 0; off >>= 1) m = fmaxf(m, __shfl_xor(m, off, 32));

  float s = 0.0f;
  for (int q = lane; q < NQ; q += 32) {
    float qv = qrow[q], sv = srow[q];
    float lp = key * qv * INV_T;
    float ln = key * (sv - qv) * INV_T;
    s += __expf(lp - m) + __expf(ln - m);
  }
#pragma unroll
  for (int off = 16; off > 0; off >>= 1) s += __shfl_xor(s, off, 32);

  if (lane == 0) {
    float lse = m + __logf(s);
    float l0  = key * qrow[0] * INV_T;           // positive logit, label 0
    atomicAdd(&per_class[k], (lse - l0) * (1.0f / CH));
  }
}

__global__ void loss_kernel(const float* __restrict__ per_class,
                            const int* __restrict__ counts,
                            float* __restrict__ out_loss) {
  float s = 0.0f;
  for (int k = 0; k < NCLS; ++k)
    if (counts[k] > 0) s += per_class[k];
  *out_loss = s;
}

extern "C" void kernel_launch(void* const* d_in, const int* in_sizes, int n_in,
                              void* d_out, int out_size, void* d_ws, size_t ws_size,
                              hipStream_t stream) {
  (void)in_sizes; (void)n_in; (void)out_size; (void)ws_size;
  const float* fea    = (const float*)d_in[0];
  const float* res    = (const float*)d_in[1];
  const float* queues = (const float*)d_in[2];
  float* out = (float*)d_out;

  char* ws = (char*)d_ws;
  int*   pred      = (int*)  (ws + 0);
  float* sums32    = (float*)(ws + 524288);
  int*   counts    = (int*)  (ws + 557056);
  float* per_class = (float*)(ws + 557184);
  float* keys      = (float*)(ws + 557312);
  float* qsum      = (float*)(ws + 576768);

  // output[0 .. BS*NCLS*HW) = res passthrough
  hipMemcpyAsync(out, res, (size_t)BS * NCLS * HW * sizeof(float),
                 hipMemcpyDeviceToDevice, stream);

  // zero sums32(8192) + counts(32) + per_class(32) = 8256 dwords (contiguous)
  zero_kernel<<<(8256 + 255) / 256, 256, 0, stream>>>(sums32, 8256);

  argmax_kernel<<<(BS * HW) / 256, 256, 0, stream>>>(res, pred, counts);
  pool_wmma_kernel<<<BS * (HW / CHUNK), 256, 0, stream>>>(fea, pred, sums32);
  keys_kernel<<<1, 256, 0, stream>>>(sums32, counts, keys);
  qsum_kernel<<<(CH * NQ + 255) / 256, 256, 0, stream>>>(queues, qsum);
  lse_kernel<<<(NCLS * CH) / 8, 256, 0, stream>>>(queues, qsum, keys, per_class);
  loss_kernel<<<1, 1, 0, stream>>>(per_class, counts,
                                   out + (size_t)BS * NCLS * HW);
}